// VIN_40973988004277
// MI455X (gfx1250) — compile-verified
//
#include <hip/hip_runtime.h>

typedef __attribute__((ext_vector_type(2))) float v2f;
typedef __attribute__((ext_vector_type(8))) float v8f;

#define IM    16
#define LQ    10
#define KITER 20          // even: final v lands in plane 1
#define HW    18          // haloed plane width (16 + 2)
#define PLANE (HW * HW)   // 324

// Optimization barrier: pin a value in a VGPR and hide its provenance so the
// compiler keeps it live across the VI loop instead of rematerializing it.
__device__ __forceinline__ void opq(int& x)   { asm("" : "+v"(x)); }
__device__ __forceinline__ void opq(float& x) { asm("" : "+v"(x)); }

// Patch index into the unified LDS plane array.
// planes: 0 = r, 1 = v_even, 2 = v_odd, 3 = zeros (K padding). vplane in {1,2}.
__device__ __forceinline__ int patch_idx(int k, int py, int px, int vplane) {
  int plane = k / 9;
  int tap   = k - plane * 9;
  int pl    = (plane == 0) ? 0 : ((plane == 1) ? vplane : 3);
  return pl * PLANE + (py + tap / 3) * HW + (px + tap % 3);
}

// Collapse the linear 2->150->1 layer pair into an effective 2->1 3x3 conv (+bias).
__global__ void vin_prep_kernel(const float* __restrict__ h_w,
                                const float* __restrict__ h_b,
                                const float* __restrict__ r_w,
                                float* __restrict__ effw) {
  int t = threadIdx.x;
  if (t < 19) {
    float acc = 0.0f;
    if (t < 18) {
      for (int c = 0; c < 150; ++c) acc += r_w[c] * h_w[c * 18 + t];
    } else {
      for (int c = 0; c < 150; ++c) acc += r_w[c] * h_b[c];
    }
    effw[t] = acc;
  }
}

// One VI sub-step for one 16-pixel tile: 5 chained f32 WMMAs + channel max.
__device__ __forceinline__ float vi_step(const float* P, const v2f A[5],
                                         const v2f B0[2], const int bidx[6],
                                         int lane) {
  v8f c = {};
  c = __builtin_amdgcn_wmma_f32_16x16x4_f32(false, A[0], false, B0[0],
                                            (short)0, c, false, false);
  c = __builtin_amdgcn_wmma_f32_16x16x4_f32(false, A[1], false, B0[1],
                                            (short)0, c, false, false);
  #pragma unroll
  for (int s = 0; s < 3; ++s) {
    v2f Bv;
    Bv.x = P[bidx[s * 2 + 0]];
    Bv.y = P[bidx[s * 2 + 1]];
    c = __builtin_amdgcn_wmma_f32_16x16x4_f32(false, A[s + 2], false, Bv,
                                              (short)0, c, false, false);
  }
  // D layout: lanes 0-15 hold M=0..7 of column n; lanes 16-31 hold M=8..15.
  float mx;
  if (lane < 16) {
    mx = fmaxf(fmaxf(fmaxf(c[0], c[1]), fmaxf(c[2], c[3])),
               fmaxf(fmaxf(c[4], c[5]), fmaxf(c[6], c[7])));
  } else {
    mx = fmaxf(c[0], c[1]);   // only M=8,9 are real channels
  }
  // Combine lane halves: ds_swizzle SWAPX16 (xor=0x10, and=0x1f)
  float other = __int_as_float(
      __builtin_amdgcn_ds_swizzle(__float_as_int(mx), 0x401f));
  return fmaxf(mx, other);
}

__global__ void __launch_bounds__(256) vin_main_kernel(
    const float* __restrict__ X,
    const float* __restrict__ S1,
    const float* __restrict__ S2,
    const float* __restrict__ q_w,
    const float* __restrict__ w_v,
    const float* __restrict__ fc_w,
    const float* __restrict__ effw_g,
    float* __restrict__ out, int Bn) {
  __shared__ float P[4 * PLANE];   // r | v_even | v_odd | zeros (haloed 18x18 each)
  __shared__ float Xs[2 * PLANE];  // haloed X, 2 channels
  __shared__ float Wl[16 * 20];    // rows 0-9 = q channels, cols 0-17 = taps, zero pad
  __shared__ float effw[20];
  __shared__ float fcw[LQ];

  const int tid = threadIdx.x;
  const int b   = blockIdx.x;

  // qw_full = concat(q_w, w) as 16x20 (M x K), zero padded for WMMA
  for (int i = tid; i < 16 * 20; i += 256) {
    int ch = i / 20, k = i % 20;
    float v = 0.0f;
    if (ch < LQ) {
      if (k < 9)       v = q_w[ch * 9 + k];
      else if (k < 18) v = w_v[ch * 9 + (k - 9)];
    }
    Wl[i] = v;
  }
  if (tid < 20) effw[tid] = (tid < 19) ? effw_g[tid] : 0.0f;
  if (tid < LQ) fcw[tid]  = fc_w[tid];

  // Zero all haloed planes: borders stay 0 forever (pad=1 convs); both v
  // planes start at 0 (iteration 0 with full weights == conv(r, q_w));
  // plane 3 permanently backs the K-padding loads.
  for (int i = tid; i < 4 * PLANE; i += 256) P[i]  = 0.0f;
  for (int i = tid; i < 2 * PLANE; i += 256) Xs[i] = 0.0f;
  __syncthreads();

  const int y = tid >> 4, x = tid & 15;
  Xs[(y + 1) * HW + (x + 1)]         = X[(size_t)b * 512 + tid];
  Xs[PLANE + (y + 1) * HW + (x + 1)] = X[(size_t)b * 512 + 256 + tid];
  __syncthreads();

  // r = conv3x3(X, effw, pad=1) + eff_bias  (collapsed hidden layer, branchless)
  {
    float acc = effw[18];
    #pragma unroll
    for (int c = 0; c < 2; ++c)
      #pragma unroll
      for (int t = 0; t < 9; ++t)
        acc += effw[c * 9 + t] * Xs[c * PLANE + (y + t / 3) * HW + (x + t % 3)];
    P[(y + 1) * HW + (x + 1)] = acc;   // plane 0 = r
  }
  __syncthreads();

  const int lane = tid & 31;
  const int wave = tid >> 5;
  const int m    = lane & 15;
  const int kh   = (lane >> 4) << 1;  // f32 16x16x4: lanes 0-15 -> K 0,1; 16-31 -> K 2,3

  // A-operands (weights), invariant across all VI iterations
  v2f A[5];
  #pragma unroll
  for (int s = 0; s < 5; ++s) {
    float ax = Wl[m * 20 + 4 * s + kh];
    float ay = Wl[m * 20 + 4 * s + kh + 1];
    opq(ax); opq(ay);
    A[s].x = ax; A[s].y = ay;
  }

  // Each wave owns two image rows (N-tiles of 16 pixels): rows 2w, 2w+1
  const int py0 = wave * 2;
  const int py1 = wave * 2 + 1;
  const int px  = m;

  // k-steps 0,1 (pure r taps): values are iteration-invariant; hoist.
  // k-steps 2-4 (v-dependent): precompute LDS indices for both v parities.
  v2f B0[2][2];
  int bE[2][6], bO[2][6];
  int wE[2], wO[2];
  #pragma unroll
  for (int tset = 0; tset < 2; ++tset) {
    int py = tset ? py1 : py0;
    #pragma unroll
    for (int s = 0; s < 2; ++s) {
      float bxv = P[patch_idx(4 * s + kh,     py, px, 1)];
      float byv = P[patch_idx(4 * s + kh + 1, py, px, 1)];
      opq(bxv); opq(byv);
      B0[tset][s].x = bxv; B0[tset][s].y = byv;
    }
    #pragma unroll
    for (int s = 2; s < 5; ++s) {
      #pragma unroll
      for (int e = 0; e < 2; ++e) {
        int ie = patch_idx(4 * s + kh + e, py, px, 1);  // reads v_even
        int io = patch_idx(4 * s + kh + e, py, px, 2);  // reads v_odd
        opq(ie); opq(io);
        bE[tset][(s - 2) * 2 + e] = ie;
        bO[tset][(s - 2) * 2 + e] = io;
      }
    }
    int we = 2 * PLANE + (py + 1) * HW + (px + 1);  // even step writes v_odd
    int wo = 1 * PLANE + (py + 1) * HW + (px + 1);  // odd step writes v_even
    opq(we); opq(wo);
    wE[tset] = we; wO[tset] = wo;
  }

  #pragma unroll 1
  for (int it = 0; it < KITER; it += 2) {
    // even sub-step: read v_even (plane 1) -> write v_odd (plane 2)
    float v0 = vi_step(P, A, B0[0], bE[0], lane);
    float v1 = vi_step(P, A, B0[1], bE[1], lane);
    if (lane < 16) { P[wE[0]] = v0; P[wE[1]] = v1; }
    __syncthreads();
    // odd sub-step: read v_odd (plane 2) -> write v_even (plane 1)
    v0 = vi_step(P, A, B0[0], bO[0], lane);
    v1 = vi_step(P, A, B0[1], bO[1], lane);
    if (lane < 16) { P[wO[0]] = v0; P[wO[1]] = v1; }
    __syncthreads();
  }

  // Final conv only matters at one gathered pixel; fc + softmax over a single
  // logit (== 1.0 identically). Final v is in plane 1 (KITER even).
  if (tid == 0) {
    int gy = (int)floorf((S1[b] + 50.0f) * (1.0f / 6.25f));
    int gx = (int)floorf((S2[b] + 50.0f) * (1.0f / 6.25f));
    gy = gy < 0 ? 0 : (gy > 15 ? 15 : gy);
    gx = gx < 0 ? 0 : (gx > 15 ? 15 : gx);
    float logit = 0.0f;
    #pragma unroll 1
    for (int ch = 0; ch < LQ; ++ch) {
      float q = 0.0f;
      #pragma unroll 1
      for (int k = 0; k < 18; ++k)
        q += Wl[ch * 20 + k] * P[patch_idx(k, gy, gx, 1)];
      logit += fcw[ch] * q;
    }
    out[b]      = logit;
    out[Bn + b] = 1.0f;
  }
}

extern "C" void kernel_launch(void* const* d_in, const int* in_sizes, int n_in,
                              void* d_out, int out_size, void* d_ws, size_t ws_size,
                              hipStream_t stream) {
  const float* X   = (const float*)d_in[0];
  const float* S1  = (const float*)d_in[1];
  const float* S2  = (const float*)d_in[2];
  const float* h_w = (const float*)d_in[3];
  const float* h_b = (const float*)d_in[4];
  const float* r_w = (const float*)d_in[5];
  const float* q_w = (const float*)d_in[6];
  const float* w_v = (const float*)d_in[7];
  const float* fcw = (const float*)d_in[8];
  float* out  = (float*)d_out;
  float* effw = (float*)d_ws;   // 19 floats of collapsed hidden-layer weights
  const int Bn = in_sizes[1];   // batch size (S1 element count)

  vin_prep_kernel<<<1, 32, 0, stream>>>(h_w, h_b, r_w, effw);
  vin_main_kernel<<<Bn, 256, 0, stream>>>(X, S1, S2, q_w, w_v, fcw, effw, out, Bn);
}